// WaveletUpsample_1563368096151
// MI455X (gfx1250) — compile-verified
//
#include <hip/hip_runtime.h>
#include <stdint.h>

typedef float v4f __attribute__((ext_vector_type(4)));
typedef int   v4i __attribute__((__vector_size__(16)));
typedef __attribute__((address_space(1))) v4i* gv4i_ptr;  // global (AS1)
typedef __attribute__((address_space(3))) v4i* lv4i_ptr;  // LDS (AS3)

#ifndef __has_builtin
#define __has_builtin(x) 0
#endif

#if __has_builtin(__builtin_amdgcn_global_store_async_from_lds_b128) && \
    __has_builtin(__builtin_amdgcn_s_wait_asynccnt)
#define USE_ASYNC_ZERO 1
#else
#define USE_ASYNC_ZERO 0
#endif

#define H_IN  256
#define W_IN  256
#define W_OUT 512

// One thread per float4 of x.
//  - dense 2*x into the top-left H x W block
//  - zero-stuffed x into even rows outside that block
//  - explicit zeros into odd rows outside that block (async-from-LDS path)
// Every output element is written exactly once:
//   reads = 128 MiB, writes = 512 MiB -> ~27 us floor at 23.3 TB/s.
__global__ __launch_bounds__(256) void wavelet_upsample(
    const float* __restrict__ x, float* __restrict__ out)
{
#if USE_ASYNC_ZERO
  __shared__ __align__(16) float zbuf[8];   // 32B of zeros for async b128 stores
  if (threadIdx.x < 8) zbuf[threadIdx.x] = 0.0f;
  __syncthreads();
#endif

  const unsigned tid = blockIdx.x * blockDim.x + threadIdx.x;
  const unsigned j4  = tid & 63u;           // float4 index along W (W/4 == 64)
  const unsigned i   = (tid >> 6) & 255u;   // row in H
  const unsigned bc  = tid >> 14;           // fused batch*channel

  const v4f* src = (const v4f*)(x + (size_t)bc * (H_IN * W_IN) + (size_t)i * W_IN) + j4;
  v4f v = __builtin_nontemporal_load(src);

  float* obase = out + (size_t)bc * (2u * H_IN * W_OUT);

  // Top-left H x W block: out[i, 4j..4j+3] = 2*x  (final value; overrides stuffing)
  v4f tv = v + v;
  __builtin_nontemporal_store(tv, (v4f*)(obase + (size_t)i * W_OUT) + j4);

  // Region NOT overwritten by the top-left block: i>=128 or col>=256 (j4>=32).
  if ((i >= 128u) | (j4 >= 32u)) {
    // Even row 2i: interleave(x, 0) over cols 8j..8j+7
    float* srow = obase + (size_t)(2u * i) * W_OUT + 8u * (size_t)j4;
    v4f s0 = {v.x, 0.0f, v.y, 0.0f};
    v4f s1 = {v.z, 0.0f, v.w, 0.0f};
    __builtin_nontemporal_store(s0, (v4f*)srow);
    __builtin_nontemporal_store(s1, (v4f*)srow + 1);

    // Odd row 2i+1: pure zeros over cols 8j..8j+7
    float* zrow = obase + (size_t)(2u * i + 1u) * W_OUT + 8u * (size_t)j4;
#if USE_ASYNC_ZERO
    // gfx1250 async store: zeros sourced from LDS, tracked by ASYNCcnt;
    // the imm offset applies to both memory and LDS address (zbuf is 32B of zeros).
    __builtin_amdgcn_global_store_async_from_lds_b128(
        (gv4i_ptr)zrow, (lv4i_ptr)&zbuf[0], 0, 0);
    __builtin_amdgcn_global_store_async_from_lds_b128(
        (gv4i_ptr)zrow, (lv4i_ptr)&zbuf[0], 16, 0);
#else
    v4f z = {0.0f, 0.0f, 0.0f, 0.0f};
    __builtin_nontemporal_store(z, (v4f*)zrow);
    __builtin_nontemporal_store(z, (v4f*)zrow + 1);
#endif
  }

#if USE_ASYNC_ZERO
  __builtin_amdgcn_s_wait_asynccnt(0);
#endif
}

extern "C" void kernel_launch(void* const* d_in, const int* in_sizes, int n_in,
                              void* d_out, int out_size, void* d_ws, size_t ws_size,
                              hipStream_t stream) {
  const float* x = (const float*)d_in[0];
  float* out = (float*)d_out;
  const int n = in_sizes[0];          // B*C*H*W = 8*64*256*256
  const int threads = n >> 2;         // one float4 per thread
  dim3 block(256);
  dim3 grid((unsigned)((threads + 255) / 256));
  wavelet_upsample<<<grid, block, 0, stream>>>(x, out);
  (void)n_in; (void)out_size; (void)d_ws; (void)ws_size;
}